// BipartiteGNN_8598524526745
// MI455X (gfx1250) — compile-verified
//
#include <hip/hip_runtime.h>
#include <hip/hip_bf16.h>

#define N_VARS  100000
#define N_CONS  50000
#define N_EDGES 1250000
#define EMB     64

typedef __attribute__((ext_vector_type(2))) float v2f;
typedef __attribute__((ext_vector_type(8))) float v8f;

// ---------------------------------------------------------------------------
// Embed: out_cat[node][0:64] = relu(feat[node] @ W + b), out_cat[node][64:128]=0
// one thread per (node, channel). 64 channels per node.
// ---------------------------------------------------------------------------
__global__ void embed_kernel(const float* __restrict__ feat, int in_dim,
                             const float* __restrict__ W,
                             const float* __restrict__ b,
                             float* __restrict__ out_cat) {
    const int tid  = blockIdx.x * blockDim.x + threadIdx.x;
    const int node = tid >> 6;
    const int j    = tid & 63;
    const float* frow = feat + (size_t)node * in_dim;
    float s = b[j];
    for (int k = 0; k < in_dim; ++k)
        s = fmaf(frow[k], W[k * EMB + j], s);
    s = s > 0.f ? s : 0.f;
    float* orow = out_cat + (size_t)node * 128;
    orow[j]      = s;     // embedding half
    orow[64 + j] = 0.f;   // aggregation half (zero every call: atomics below)
}

// ---------------------------------------------------------------------------
// Edge scatter: dst[dst_idx[e]][col_off + j] += src[src_idx[e]][j] * ea[e]*we[j]
// one thread per (edge, channel): coalesced rows, f32 atomics land in L2.
// ---------------------------------------------------------------------------
__global__ void scatter_kernel(const float* __restrict__ src_emb, int src_stride,
                               const int* __restrict__ gather_idx,
                               const int* __restrict__ scatter_idx,
                               const float* __restrict__ edge_attr,
                               const float* __restrict__ w_edge,
                               float* __restrict__ dst, int dst_stride, int col_off) {
    const long long tid = (long long)blockIdx.x * blockDim.x + threadIdx.x;
    const int e = (int)(tid >> 6);
    const int j = (int)(tid & 63);
    const int gs = gather_idx[e];
    const int ds = scatter_idx[e];
    const float val = src_emb[(size_t)gs * src_stride + j] * edge_attr[e] * w_edge[j];
    atomicAdd(&dst[(size_t)ds * dst_stride + col_off + j], val);
}

// ---------------------------------------------------------------------------
// WMMA fp32 GEMM: out[M x 64] = relu(A[M x K] @ W[K x 64] + b)
// one wave (32 lanes) per 16-row tile; 4 C-tiles cover all 64 output columns.
// V_WMMA_F32_16X16X4_F32: A 16x4, B 4x16, C 16x16 (exact fp32 path).
// ---------------------------------------------------------------------------
__global__ void __launch_bounds__(32)
gemm_relu_kernel(const float* __restrict__ A, int K,
                 const float* __restrict__ W,
                 const float* __restrict__ b,
                 float* __restrict__ out, int out_stride) {
    const int lane = threadIdx.x & 31;
    const int half = lane >> 4;     // 0: rows v / K even-pair, 1: rows v+8 / K odd-pair
    const int l    = lane & 15;
    const int m0   = blockIdx.x * 16;

    v8f acc[4];
#pragma unroll
    for (int ct = 0; ct < 4; ++ct) {
        const float bv = b[16 * ct + l];   // bias depends on N only
#pragma unroll
        for (int v = 0; v < 8; ++v) acc[ct][v] = bv;
    }

    const float* arow = A + (size_t)(m0 + l) * K;
    for (int k0 = 0; k0 < K; k0 += 4) {
        const int ka = k0 + 2 * half;
        v2f a;                               // A frag: lane l holds row m0+l, K=ka,ka+1
        a.x = arow[ka];
        a.y = arow[ka + 1];
#pragma unroll
        for (int ct = 0; ct < 4; ++ct) {
            const int n = 16 * ct + l;
            v2f bb;                          // B frag: K=ka,ka+1 striped over N=l
            bb.x = W[(size_t)ka * EMB + n];
            bb.y = W[(size_t)(ka + 1) * EMB + n];
            acc[ct] = __builtin_amdgcn_wmma_f32_16x16x4_f32(
                false, a, false, bb, (short)0, acc[ct], false, false);
        }
    }

#pragma unroll
    for (int ct = 0; ct < 4; ++ct) {
#pragma unroll
        for (int v = 0; v < 8; ++v) {
            float x = acc[ct][v];
            x = x > 0.f ? x : 0.f;
            out[(size_t)(m0 + v + 8 * half) * out_stride + 16 * ct + l] = x;
        }
    }
}

// ---------------------------------------------------------------------------
// Head: scores = relu(A @ W1 + b1) @ W2 + b2   (K = 64), fused reduction.
// WMMA for the 64x64 GEMM, then per-half-wave shfl_xor butterfly for the
// 64 -> 1 projection (lanes 0-15 hold rows v, lanes 16-31 rows v+8).
// ---------------------------------------------------------------------------
__global__ void __launch_bounds__(32)
head_kernel(const float* __restrict__ A,
            const float* __restrict__ W1, const float* __restrict__ b1,
            const float* __restrict__ W2, const float* __restrict__ b2,
            float* __restrict__ out) {
    const int lane = threadIdx.x & 31;
    const int half = lane >> 4;
    const int l    = lane & 15;
    const int m0   = blockIdx.x * 16;

    v8f acc[4];
#pragma unroll
    for (int ct = 0; ct < 4; ++ct) {
        const float bv = b1[16 * ct + l];
#pragma unroll
        for (int v = 0; v < 8; ++v) acc[ct][v] = bv;
    }

    const float* arow = A + (size_t)(m0 + l) * EMB;
    for (int k0 = 0; k0 < EMB; k0 += 4) {
        const int ka = k0 + 2 * half;
        v2f a;
        a.x = arow[ka];
        a.y = arow[ka + 1];
#pragma unroll
        for (int ct = 0; ct < 4; ++ct) {
            const int n = 16 * ct + l;
            v2f bb;
            bb.x = W1[(size_t)ka * EMB + n];
            bb.y = W1[(size_t)(ka + 1) * EMB + n];
            acc[ct] = __builtin_amdgcn_wmma_f32_16x16x4_f32(
                false, a, false, bb, (short)0, acc[ct], false, false);
        }
    }

    // relu(h) . W2, accumulated per lane then reduced over each 16-lane half
    float s[8];
#pragma unroll
    for (int v = 0; v < 8; ++v) s[v] = 0.f;
#pragma unroll
    for (int ct = 0; ct < 4; ++ct) {
        const float w2 = W2[16 * ct + l];
#pragma unroll
        for (int v = 0; v < 8; ++v) {
            float x = acc[ct][v];
            x = x > 0.f ? x : 0.f;
            s[v] = fmaf(x, w2, s[v]);
        }
    }
#pragma unroll
    for (int v = 0; v < 8; ++v) {
        float r = s[v];
        r += __shfl_xor(r, 1, 32);
        r += __shfl_xor(r, 2, 32);
        r += __shfl_xor(r, 4, 32);
        r += __shfl_xor(r, 8, 32);   // xor<16 stays within each half-wave
        s[v] = r;
    }
    if (l == 0) {
        const float bp2 = b2[0];
#pragma unroll
        for (int v = 0; v < 8; ++v)
            out[m0 + v + 8 * half] = s[v] + bp2;  // half=0 -> rows v, half=1 -> rows v+8
    }
}

// ---------------------------------------------------------------------------
extern "C" void kernel_launch(void* const* d_in, const int* in_sizes, int n_in,
                              void* d_out, int out_size, void* d_ws, size_t ws_size,
                              hipStream_t stream) {
    const float* var_features = (const float*)d_in[0];
    const float* con_features = (const float*)d_in[1];
    const int*   edge_index   = (const int*)d_in[2];   // [2, N_EDGES]: row0=con, row1=var
    const float* edge_attr    = (const float*)d_in[3];
    const float* W_var = (const float*)d_in[4];
    const float* b_var = (const float*)d_in[5];
    const float* W_con = (const float*)d_in[6];
    const float* b_con = (const float*)d_in[7];
    const float* w_edge = (const float*)d_in[8];
    const float* W_cu  = (const float*)d_in[9];
    const float* b_cu  = (const float*)d_in[10];
    const float* W_vu  = (const float*)d_in[11];
    const float* b_vu  = (const float*)d_in[12];
    const float* W_p1  = (const float*)d_in[13];
    const float* b_p1  = (const float*)d_in[14];
    const float* W_p2  = (const float*)d_in[15];
    const float* b_p2  = (const float*)d_in[16];
    float* out = (float*)d_out;

    const int* con_idx = edge_index;            // edge_index[0]
    const int* var_idx = edge_index + N_EDGES;  // edge_index[1]

    // workspace layout (256B-aligned offsets)
    char* ws = (char*)d_ws;
    float* var_cat  = (float*)(ws);              // [N_VARS, 128]  (emb | agg)
    float* con_cat  = (float*)(ws + 51200000);   // [N_CONS, 128]  (emb | agg)
    float* con_emb1 = (float*)(ws + 76800000);   // [N_CONS, 64]
    float* var_emb2 = (float*)(ws + 89600000);   // [N_VARS, 64]

    // 1) node embeddings (also zeroes agg halves -> replay-safe)
    embed_kernel<<<(N_VARS * 64) / 256, 256, 0, stream>>>(var_features, 19, W_var, b_var, var_cat);
    embed_kernel<<<(N_CONS * 64) / 256, 256, 0, stream>>>(con_features, 5, W_con, b_con, con_cat);

    // 2) var -> con messages into con_cat[:,64:128]
    scatter_kernel<<<(int)(((long long)N_EDGES * 64) / 256), 256, 0, stream>>>(
        var_cat, 128, var_idx, con_idx, edge_attr, w_edge, con_cat, 128, 64);

    // 3) con update: relu([con_emb | con_agg] @ W_cu + b_cu)   (WMMA, K=128)
    gemm_relu_kernel<<<N_CONS / 16, 32, 0, stream>>>(con_cat, 128, W_cu, b_cu, con_emb1, 64);

    // 4) con -> var messages into var_cat[:,64:128]
    scatter_kernel<<<(int)(((long long)N_EDGES * 64) / 256), 256, 0, stream>>>(
        con_emb1, 64, con_idx, var_idx, edge_attr, w_edge, var_cat, 128, 64);

    // 5) var update: relu([var_emb | var_agg] @ W_vu + b_vu)   (WMMA, K=128)
    gemm_relu_kernel<<<N_VARS / 16, 32, 0, stream>>>(var_cat, 128, W_vu, b_vu, var_emb2, 64);

    // 6) head: relu(var_emb2 @ W_p1 + b_p1) @ W_p2 + b_p2      (WMMA + fused reduce)
    head_kernel<<<N_VARS / 16, 32, 0, stream>>>(var_emb2, W_p1, b_p1, W_p2, b_p2, out);
}